// NoisyTopkRouter_63015760167498
// MI455X (gfx1250) — compile-verified
//
#include <hip/hip_runtime.h>
#include <math.h>

#define D_EMBED   2048
#define N_EXP     64
#define TOPK      8
#define ROWS      64      // token rows per block
#define BK        32      // K-chunk (shrunk so 2x double-buffered tiles fit LDS)
#define NCHUNK    (D_EMBED / BK)   // 64
#define XS_STRIDE 36      // 32 + 4 pad  -> A reads conflict-free (4*9 pattern)
#define WS_STRIDE 136     // 128 + 8 pad -> B reads conflict-free
#define MS_STRIDE 65      // epilogue row stride

#define AS1 __attribute__((address_space(1)))
#define AS3 __attribute__((address_space(3)))

#if __has_builtin(__builtin_amdgcn_global_load_async_to_lds_b128) && \
    __has_builtin(__builtin_amdgcn_s_wait_asynccnt)
#define USE_ASYNC 1
#else
#define USE_ASYNC 0
#endif

typedef float v2f __attribute__((ext_vector_type(2)));
typedef float v8f __attribute__((ext_vector_type(8)));
typedef int   b128_t __attribute__((vector_size(16)));   // builtin's expected pointee

union SharedU {
    struct { float Xs[2][ROWS][XS_STRIDE]; float Ws[2][BK][WS_STRIDE]; } g; // 53248 B
    struct { float Lg[ROWS][MS_STRIDE];    float Sc[ROWS][MS_STRIDE];   } e; // 33280 B
};

__device__ __forceinline__ float softplus_f(float z) {
    return fmaxf(z, 0.0f) + log1pf(expf(-fabsf(z)));
}

// Stage one K-chunk of x (64x32) and fused [W1|W2] (32x128) into LDS buffer `buf`.
__device__ __forceinline__ void stage_chunk(SharedU& sh, int buf, int kc, int tid,
                                            int rowBase,
                                            const float* __restrict__ x,
                                            const float* __restrict__ W1,
                                            const float* __restrict__ W2)
{
    // ---- x tile: 2048 floats = 512 float4, 2 per thread (coalesced)
    #pragma unroll
    for (int j = 0; j < 2; ++j) {
        int fi4 = tid + 256 * j;              // 0..511
        int r = fi4 >> 3;                     // (fi4*4) / 32
        int c = (fi4 & 7) << 2;
        const float* src = &x[(size_t)(rowBase + r) * D_EMBED + kc * BK + c];
        float* dst = &sh.g.Xs[buf][r][c];
#if USE_ASYNC
        __builtin_amdgcn_global_load_async_to_lds_b128((AS1 b128_t*)src, (AS3 b128_t*)dst, 0, 0);
#else
        *(float4*)dst = *(const float4*)src;
#endif
    }
    // ---- W tile: 4096 floats = 1024 float4, 4 per thread
    #pragma unroll
    for (int j = 0; j < 4; ++j) {
        int fi4 = tid + 256 * j;              // 0..1023
        int k = fi4 >> 5;                     // (fi4*4) / 128
        int n = (fi4 & 31) << 2;              // 0..124, never crosses the 64 boundary
        const float* src = (n < N_EXP)
            ? &W1[(size_t)(kc * BK + k) * N_EXP + n]
            : &W2[(size_t)(kc * BK + k) * N_EXP + (n - N_EXP)];
        float* dst = &sh.g.Ws[buf][k][n];
#if USE_ASYNC
        __builtin_amdgcn_global_load_async_to_lds_b128((AS1 b128_t*)src, (AS3 b128_t*)dst, 0, 0);
#else
        *(float4*)dst = *(const float4*)src;
#endif
    }
}

__global__ __launch_bounds__(256, 2)
void noisy_topk_router_kernel(const float* __restrict__ x,
                              const float* __restrict__ noise,
                              const float* __restrict__ W1,
                              const float* __restrict__ b1,
                              const float* __restrict__ W2,
                              const float* __restrict__ b2,
                              float* __restrict__ outSparse,
                              float* __restrict__ outIdx,
                              float* __restrict__ outFull,
                              int T)
{
    __shared__ SharedU sh;

    const int tid  = threadIdx.x;
    const int wave = tid >> 5;
    const int lane = tid & 31;
    const int rowBase = blockIdx.x * ROWS;
    if (rowBase >= T) return;

    const int n0 = (wave & 3) * 16;   // 16-expert column slice of this wave
    const int m0 = lane & 15;
    const int hiGrp = lane >> 4;      // 0: lanes 0-15, 1: lanes 16-31
    const int koff = hiGrp << 1;

    v8f acc[4];
    #pragma unroll
    for (int mt = 0; mt < 4; ++mt)
        #pragma unroll
        for (int v = 0; v < 8; ++v) acc[mt][v] = 0.0f;

#if USE_ASYNC
    stage_chunk(sh, 0, 0, tid, rowBase, x, W1, W2);   // prologue: chunk 0 in flight
#endif

    for (int kc = 0; kc < NCHUNK; ++kc) {
        const int cur = kc & 1;
#if USE_ASYNC
        __builtin_amdgcn_s_wait_asynccnt(0);  // my async writes for chunk kc done
        __syncthreads();                      // everyone's writes visible; prev reads done
        if (kc + 1 < NCHUNK)
            stage_chunk(sh, cur ^ 1, kc + 1, tid, rowBase, x, W1, W2);
#else
        __syncthreads();                      // prev compute reads done
        stage_chunk(sh, cur, kc, tid, rowBase, x, W1, W2);
        if (kc + 1 < NCHUNK) {
            int r = tid >> 3;
            int c = (tid & 7) << 2;
            __builtin_prefetch(&x[(size_t)(rowBase + r) * D_EMBED + (kc + 1) * BK + c], 0, 1);
        }
        __syncthreads();
#endif
        const float (*Xs)[XS_STRIDE] = sh.g.Xs[cur];
        const float (*Ws)[WS_STRIDE] = sh.g.Ws[cur];

#if __has_builtin(__builtin_amdgcn_wmma_f32_16x16x4_f32)
        // fp32 matrix pipe: D = A(16x4) * B(4x16) + C, K-step 4
        #pragma unroll
        for (int k4 = 0; k4 < BK / 4; ++k4) {
            const int k0 = k4 * 4;
            v2f b;                            // lanes 0-15: K=k0,k0+1 ; lanes 16-31: K=k0+2,k0+3
            b.x = Ws[k0 + koff][n0 + m0];
            b.y = Ws[k0 + koff + 1][n0 + m0];
            #pragma unroll
            for (int mt = 0; mt < 4; ++mt) {
                v2f a = *(const v2f*)&Xs[mt * 16 + m0][k0 + koff];
                acc[mt] = __builtin_amdgcn_wmma_f32_16x16x4_f32(
                    false, a, false, b, (short)0, acc[mt], false, false);
            }
        }
#else
        // VALU fallback with identical accumulator layout
        for (int k = 0; k < BK; ++k) {
            float bv = Ws[k][n0 + m0];
            #pragma unroll
            for (int mt = 0; mt < 4; ++mt)
                #pragma unroll
                for (int v = 0; v < 8; ++v)
                    acc[mt][v] = fmaf(Xs[mt * 16 + v + 8 * hiGrp][k], bv, acc[mt][v]);
        }
#endif
    }

    // ---- epilogue: spill accumulators to LDS (reuse tile memory via union)
#if USE_ASYNC
    __builtin_amdgcn_s_wait_asynccnt(0);
#endif
    __syncthreads();
    {
        const int colIn = n0 + m0;
        if (wave < 4) {
            float bb = b1[colIn];
            #pragma unroll
            for (int mt = 0; mt < 4; ++mt)
                #pragma unroll
                for (int v = 0; v < 8; ++v)
                    sh.e.Lg[mt * 16 + v + 8 * hiGrp][colIn] = acc[mt][v] + bb;
        } else {
            float bb = b2[colIn];
            #pragma unroll
            for (int mt = 0; mt < 4; ++mt)
                #pragma unroll
                for (int v = 0; v < 8; ++v)
                    sh.e.Sc[mt * 16 + v + 8 * hiGrp][colIn] = softplus_f(acc[mt][v] + bb);
        }
    }
    __syncthreads();

    // ---- per-row softmax / top-8 / masked softmax (one thread per token row)
    if (tid < ROWS) {
        const int r = tid;
        const size_t gRow = (size_t)(rowBase + r);
        const float* nz = &noise[gRow * N_EXP];
        float* Mrow = sh.e.Lg[r];
        const float* Srow = sh.e.Sc[r];

        float vmax = -INFINITY;
        for (int e = 0; e < N_EXP; ++e) {
            float m = fmaf(nz[e], Srow[e], Mrow[e]);   // logits + noise*scale
            Mrow[e] = m;
            vmax = fmaxf(vmax, m);
        }
        // full softmax
        float sum = 0.0f;
        for (int e = 0; e < N_EXP; ++e) sum += expf(Mrow[e] - vmax);
        float inv = 1.0f / sum;
        float* fo = &outFull[gRow * N_EXP];
        for (int e = 0; e < N_EXP; ++e) fo[e] = expf(Mrow[e] - vmax) * inv;

        // top-8, descending, lowest-index on ties (strict >)
        float tv[TOPK]; int ti[TOPK];
        #pragma unroll
        for (int k = 0; k < TOPK; ++k) {
            float best = -INFINITY; int bi = 0;
            for (int e = 0; e < N_EXP; ++e) {
                float m = Mrow[e];
                if (m > best) { best = m; bi = e; }
            }
            tv[k] = best; ti[k] = bi;
            Mrow[bi] = -INFINITY;
        }
        // masked (sparse) softmax over the top-8
        float tsum = 0.0f;
        #pragma unroll
        for (int k = 0; k < TOPK; ++k) tsum += expf(tv[k] - tv[0]);
        float tinv = 1.0f / tsum;
        float* sp = &outSparse[gRow * N_EXP];
        for (int e = 0; e < N_EXP; ++e) sp[e] = 0.0f;
        #pragma unroll
        for (int k = 0; k < TOPK; ++k) sp[ti[k]] = expf(tv[k] - tv[0]) * tinv;
        #pragma unroll
        for (int k = 0; k < TOPK; ++k) outIdx[gRow * TOPK + k] = (float)ti[k];
    }
}

extern "C" void kernel_launch(void* const* d_in, const int* in_sizes, int n_in,
                              void* d_out, int out_size, void* d_ws, size_t ws_size,
                              hipStream_t stream) {
    const float* x     = (const float*)d_in[0];
    const float* noise = (const float*)d_in[1];
    const float* W1    = (const float*)d_in[2];
    const float* b1    = (const float*)d_in[3];
    const float* W2    = (const float*)d_in[4];
    const float* b2    = (const float*)d_in[5];

    const int T = in_sizes[0] / D_EMBED;   // 16384 tokens

    float* out       = (float*)d_out;
    float* outSparse = out;                              // [T, 64]
    float* outIdx    = out + (size_t)T * N_EXP;          // [T, 8]
    float* outFull   = outIdx + (size_t)T * TOPK;        // [T, 64]

    const int blocks = T / ROWS;
    hipLaunchKernelGGL(noisy_topk_router_kernel, dim3(blocks), dim3(256), 0, stream,
                       x, noise, W1, b1, W2, b2, outSparse, outIdx, outFull, T);
}